// HGNNConv_61649960566909
// MI455X (gfx1250) — compile-verified
//
#include <hip/hip_runtime.h>

typedef __attribute__((ext_vector_type(2))) float v2f;
typedef __attribute__((ext_vector_type(8))) float v8f;

#define C_DIM 128
#define LDS_STRIDE 132  // pad 128 -> 132 so A-fragment reads are bank-conflict-free

// ---------------------------------------------------------------------------
// xt[N x 128] = X[N x 128] @ W[128 x 128], fp32, via V_WMMA_F32_16X16X4_F32.
// 256 threads = 8 wave32s per block; block owns 16 rows, wave w owns cols
// [16w, 16w+16). K=128 processed as 4 LDS-staged chunks of 32 (8 WMMA each).
// Requires N % 16 == 0 (N = 100000 = 6250*16).
// ---------------------------------------------------------------------------
__global__ __launch_bounds__(256) void gemm128_wmma_f32(
    const float* __restrict__ X, const float* __restrict__ W,
    float* __restrict__ XT) {
  __shared__ float sX[16 * LDS_STRIDE];
  __shared__ float sW[32 * LDS_STRIDE];
  const int t    = threadIdx.x;
  const int wave = t >> 5;
  const int lane = t & 31;
  const int r0   = blockIdx.x * 16;
  const int c0   = wave * 16;
  const int hi   = lane >> 4;   // 0: lanes 0-15, 1: lanes 16-31
  const int l    = lane & 15;

  // stage the 16x128 X strip (512 float4 / 256 threads)
  for (int v = 0; v < 2; ++v) {
    int fi  = t + v * 256;            // float4 index
    int row = fi >> 5;                // 32 float4 per row
    int c4  = fi & 31;
    const float4 g = ((const float4*)(X + (size_t)(r0 + row) * C_DIM))[c4];
    float* d = &sX[row * LDS_STRIDE + c4 * 4];
    d[0] = g.x; d[1] = g.y; d[2] = g.z; d[3] = g.w;
  }

  v8f acc = {};
  for (int kb = 0; kb < 4; ++kb) {
    __syncthreads();
    // stage 32x128 chunk of W (1024 float4 / 256 threads)
    for (int v = 0; v < 4; ++v) {
      int fi  = t + v * 256;
      int row = fi >> 5;
      int c4  = fi & 31;
      const float4 g = ((const float4*)(W + (size_t)(kb * 32 + row) * C_DIM))[c4];
      float* d = &sW[row * LDS_STRIDE + c4 * 4];
      d[0] = g.x; d[1] = g.y; d[2] = g.z; d[3] = g.w;
    }
    __syncthreads();
#pragma unroll
    for (int ks = 0; ks < 8; ++ks) {
      const int k  = kb * 32 + ks * 4;  // global K within the strip
      const int kl = ks * 4;            // K within the staged W chunk
      v2f a, b;
      // A 16x4 fp32 layout (ISA 7.12.2): lanes 0-15 -> K=0,1 ; lanes 16-31 -> K=2,3
      a.x = sX[l * LDS_STRIDE + k + (hi ? 2 : 0)];
      a.y = sX[l * LDS_STRIDE + k + (hi ? 3 : 1)];
      // B 4x16 fp32 layout (mirrored): v0 rows K=0/2, v1 rows K=1/3, N striped on lanes
      b.x = sW[(kl + (hi ? 2 : 0)) * LDS_STRIDE + c0 + l];
      b.y = sW[(kl + (hi ? 3 : 1)) * LDS_STRIDE + c0 + l];
      acc = __builtin_amdgcn_wmma_f32_16x16x4_f32(
          /*neg_a=*/false, a, /*neg_b=*/false, b,
          /*c_mod=*/(short)0, acc, /*reuse_a=*/false, /*reuse_b=*/false);
    }
  }
  // C/D 16x16 fp32 layout: VGPR j = row (j + 8*hi), col = lane&15
  float* out = XT + (size_t)r0 * C_DIM + c0;
#pragma unroll
  for (int j = 0; j < 8; ++j) {
    out[(size_t)(j + (hi ? 8 : 0)) * C_DIM + l] = acc[j];
  }
}

// ---------------------------------------------------------------------------
// helpers
// ---------------------------------------------------------------------------
__global__ void zero_f4(float4* __restrict__ p, long n4) {
  long i = (long)blockIdx.x * blockDim.x + threadIdx.x;
  if (i < n4) p[i] = make_float4(0.f, 0.f, 0.f, 0.f);
}

// D[n] += hw[e], Bc[e] += 1 over all incidences
__global__ void degree_kernel(const int* __restrict__ nidx, const int* __restrict__ eidx,
                              const float* __restrict__ hw,
                              float* __restrict__ D, float* __restrict__ Bc, int E) {
  int i = blockIdx.x * blockDim.x + threadIdx.x;
  if (i < E) {
    int e = eidx[i];
    atomicAdd(&D[nidx[i]], hw[e]);
    atomicAdd(&Bc[e], 1.0f);
  }
}

// in-place safe reciprocal: v -> (v>0) ? 1/v : 0
__global__ void inv_kernel(float* __restrict__ p, int n) {
  int i = blockIdx.x * blockDim.x + threadIdx.x;
  if (i < n) {
    float v = p[i];
    p[i] = (v > 0.f) ? (1.0f / v) : 0.0f;
  }
}

// node -> hyperedge: e[ei] += Binv[ei] * xt[ni]   (one wave32 per incidence,
// each lane owns a float4 of the 128-wide row; fp32 global atomics)
__global__ __launch_bounds__(256) void scatter_n2e(
    const float* __restrict__ xt, const int* __restrict__ nidx,
    const int* __restrict__ eidx, const float* __restrict__ Binv,
    float* __restrict__ ebuf, int E) {
  int wi   = (int)((blockIdx.x * (unsigned)blockDim.x + threadIdx.x) >> 5);
  int lane = threadIdx.x & 31;
  if (wi >= E) return;
  int n = nidx[wi];
  int e = eidx[wi];
  float s = Binv[e];
  float4 v = ((const float4*)(xt + (size_t)n * C_DIM))[lane];
  float* dst = ebuf + (size_t)e * C_DIM + lane * 4;
  atomicAdd(dst + 0, s * v.x);
  atomicAdd(dst + 1, s * v.y);
  atomicAdd(dst + 2, s * v.z);
  atomicAdd(dst + 3, s * v.w);
}

// hyperedge -> node: acc[ni] += Dinv[ni] * e[ei]
__global__ __launch_bounds__(256) void scatter_e2n(
    const float* __restrict__ ebuf, const int* __restrict__ nidx,
    const int* __restrict__ eidx, const float* __restrict__ Dinv,
    float* __restrict__ acc, int E) {
  int wi   = (int)((blockIdx.x * (unsigned)blockDim.x + threadIdx.x) >> 5);
  int lane = threadIdx.x & 31;
  if (wi >= E) return;
  int n = nidx[wi];
  int e = eidx[wi];
  float s = Dinv[n];
  float4 v = ((const float4*)(ebuf + (size_t)e * C_DIM))[lane];
  float* dst = acc + (size_t)n * C_DIM + lane * 4;
  atomicAdd(dst + 0, s * v.x);
  atomicAdd(dst + 1, s * v.y);
  atomicAdd(dst + 2, s * v.z);
  atomicAdd(dst + 3, s * v.w);
}

// out = relu(acc + bias), float4-vectorized; row stride 128 floats = 32 float4
__global__ void bias_relu(const float4* __restrict__ acc, const float4* __restrict__ bias,
                          float4* __restrict__ out, long n4) {
  long i = (long)blockIdx.x * blockDim.x + threadIdx.x;
  if (i < n4) {
    float4 a = acc[i];
    float4 b = bias[i & 31];
    float4 r;
    r.x = fmaxf(a.x + b.x, 0.f);
    r.y = fmaxf(a.y + b.y, 0.f);
    r.z = fmaxf(a.z + b.z, 0.f);
    r.w = fmaxf(a.w + b.w, 0.f);
    out[i] = r;
  }
}

// ---------------------------------------------------------------------------
// launch
// ---------------------------------------------------------------------------
extern "C" void kernel_launch(void* const* d_in, const int* in_sizes, int n_in,
                              void* d_out, int out_size, void* d_ws, size_t ws_size,
                              hipStream_t stream) {
  const float* x    = (const float*)d_in[0];
  const int*   hidx = (const int*)d_in[1];
  const float* hw   = (const float*)d_in[2];
  const float* Wm[3] = {(const float*)d_in[5], (const float*)d_in[7], (const float*)d_in[9]};
  const float* bv[3] = {(const float*)d_in[6], (const float*)d_in[8], (const float*)d_in[10]};

  const int N = in_sizes[0] / C_DIM;   // 100000 (multiple of 16)
  const int E = in_sizes[1] / 2;       // 1600000
  const int M = in_sizes[2];           // 20000
  const int* nidx = hidx;              // hyperedge_index[0]
  const int* eidx = hidx + E;          // hyperedge_index[1]

  // workspace carve-up (all fp32)
  float* ws    = (float*)d_ws;
  size_t off   = 0;
  float* Dinv  = ws + off; off += (size_t)N;
  float* Binv  = ws + off; off += (size_t)M;
  float* xt    = ws + off; off += (size_t)N * C_DIM;
  float* ebuf  = ws + off; off += (size_t)M * C_DIM;
  float* accb  = ws + off; off += (size_t)N * C_DIM;
  float* hbuf  = ws + off; off += (size_t)N * C_DIM;

  const long nN4  = (long)N * C_DIM / 4;
  const long nM4  = (long)M * C_DIM / 4;
  const int  TB   = 256;

  // ---- layer-invariant degree normalizers ----
  zero_f4<<<(unsigned)((N / 4 + TB - 1) / TB), TB, 0, stream>>>((float4*)Dinv, N / 4);
  zero_f4<<<(unsigned)((M / 4 + TB - 1) / TB), TB, 0, stream>>>((float4*)Binv, M / 4);
  degree_kernel<<<(unsigned)((E + TB - 1) / TB), TB, 0, stream>>>(nidx, eidx, hw, Dinv, Binv, E);
  inv_kernel<<<(unsigned)((N + TB - 1) / TB), TB, 0, stream>>>(Dinv, N);
  inv_kernel<<<(unsigned)((M + TB - 1) / TB), TB, 0, stream>>>(Binv, M);

  const unsigned scatterGrid = (unsigned)(((long)E * 32 + TB - 1) / TB);  // wave per edge
  const float* cur = x;
  for (int L = 0; L < 3; ++L) {
    float* outL = (L == 2) ? (float*)d_out : hbuf;
    // xt = cur @ W[L]   (WMMA fp32)
    gemm128_wmma_f32<<<(unsigned)(N / 16), TB, 0, stream>>>(cur, Wm[L], xt);
    // e = segsum_edge(Binv[e] * xt[n])
    zero_f4<<<(unsigned)((nM4 + TB - 1) / TB), TB, 0, stream>>>((float4*)ebuf, nM4);
    scatter_n2e<<<scatterGrid, TB, 0, stream>>>(xt, nidx, eidx, Binv, ebuf, E);
    // acc = segsum_node(Dinv[n] * e[e])
    zero_f4<<<(unsigned)((nN4 + TB - 1) / TB), TB, 0, stream>>>((float4*)accb, nN4);
    scatter_e2n<<<scatterGrid, TB, 0, stream>>>(ebuf, nidx, eidx, Dinv, accb, E);
    // out = relu(acc + b)
    bias_relu<<<(unsigned)((nN4 + TB - 1) / TB), TB, 0, stream>>>(
        (const float4*)accb, (const float4*)bv[L], (float4*)outL, nN4);
    cur = outL;
  }
}